// LSTM_77180562309327
// MI455X (gfx1250) — compile-verified
//
#include <hip/hip_runtime.h>
#include <math.h>

typedef float v2f __attribute__((ext_vector_type(2)));
typedef float v8f __attribute__((ext_vector_type(8)));

#define HDIM 256
#define SEQ  1024
#define NT   16    // hidden n-tiles of 16
#define KS   64    // K steps of 4 (K = 256)
#define LSTR 260   // padded LDS row stride in dwords (260 % 64 == 4 -> conflict-free)

// ---------------------------------------------------------------------------
// One-time repack of the four recurrent weights (H x H, row-major) into
// WMMA-B operand order so the hot loop does one coalesced float2 load per
// lane per (gate, kstep).  B 4x16 layout: lane l supplies N = l&15,
// K = k0 + (l>>4)*2 + v for v in {0,1}.
// ws float layout: (((g*NT + ntile)*KS + ks)*32 + lane)*2 + v   (1 MB total)
// ---------------------------------------------------------------------------
__global__ __launch_bounds__(256)
void lstm_repack(const float* __restrict__ Wgh, const float* __restrict__ Wih,
                 const float* __restrict__ Wfh, const float* __restrict__ Woh,
                 float* __restrict__ ws) {
    int idx = blockIdx.x * 256 + threadIdx.x;          // 0 .. 2^17-1
    if (idx >= 4 * NT * KS * 32) return;
    int lane = idx & 31;
    int ks   = (idx >> 5) & 63;
    int t    = (idx >> 11) & 15;
    int g    = idx >> 15;
    const float* W = (g == 0) ? Wgh : (g == 1) ? Wih : (g == 2) ? Wfh : Woh;
    int k = ks * 4 + (lane >> 4) * 2;
    int n = t * 16 + (lane & 15);
    ws[idx * 2 + 0] = W[(k + 0) * HDIM + n];
    ws[idx * 2 + 1] = W[(k + 1) * HDIM + n];
}

// Branch-free activations on native TRANS ops (v_exp_f32 / v_rcp_f32).
// tanh(x)    = 1 - 2 / (exp2(x * 2*log2(e)) + 1)     (saturates correctly)
// sigmoid(x) = 1 / (1 + exp2(-x * log2(e)))
__device__ __forceinline__ float fast_tanh(float x) {
    float e = __builtin_amdgcn_exp2f(x * 2.885390081777927f);
    return 1.0f - 2.0f * __builtin_amdgcn_rcpf(e + 1.0f);
}
__device__ __forceinline__ float fast_sigmoid(float x) {
    float e = __builtin_amdgcn_exp2f(x * -1.4426950408889634f);
    return __builtin_amdgcn_rcpf(1.0f + e);
}

// ---------------------------------------------------------------------------
// Persistent per-block LSTM: block b owns batches [16b, 16b+16).  Batches are
// independent through the recurrence, so there is NO cross-block traffic.
// Hidden state h lives in LDS; cell state c lives in the owning wave's VGPRs.
// Wave w owns hidden columns [16w, 16w+16) and all four gates for them.
// ---------------------------------------------------------------------------
__global__ __launch_bounds__(512)
void lstm_main(const float* __restrict__ x,   // (B, S)
               const float* __restrict__ ws,  // repacked Wh, 1 MB
               const float* __restrict__ Wgx, const float* __restrict__ b_gx,
               const float* __restrict__ b_gh,
               const float* __restrict__ Wix, const float* __restrict__ b_ix,
               const float* __restrict__ b_ih,
               const float* __restrict__ Wfx, const float* __restrict__ b_fx,
               const float* __restrict__ b_fh,
               const float* __restrict__ Wox, const float* __restrict__ b_ox,
               const float* __restrict__ b_oh,
               const float* __restrict__ Wp,  const float* __restrict__ b_p,
               float* __restrict__ out) {
    __shared__ float hls[16 * LSTR];          // h[m][k], padded rows

    const int tid  = threadIdx.x;
    const int wave = tid >> 5;                // n-tile 0..15
    const int lane = tid & 31;
    const int m0   = blockIdx.x * 16;         // batch tile base
    const int n0   = wave * 16;
    const int nn   = lane & 15;               // column within tile (B/C/D lane)
    const int hi   = lane >> 4;               // lane half (A/B K select, C/D M+8)

    // zero-init hidden state
    for (int i = tid; i < 16 * LSTR; i += 512) hls[i] = 0.0f;

    // per-column constants: x-weight and fused biases per gate
    const float wx0 = Wgx[n0 + nn], bb0 = b_gx[n0 + nn] + b_gh[n0 + nn];
    const float wx1 = Wix[n0 + nn], bb1 = b_ix[n0 + nn] + b_ih[n0 + nn];
    const float wx2 = Wfx[n0 + nn], bb2 = b_fx[n0 + nn] + b_fh[n0 + nn];
    const float wx3 = Wox[n0 + nn], bb3 = b_ox[n0 + nn] + b_oh[n0 + nn];

    // A operand base: row m = lane&15, cols k0 + (lane>>4)*2 + {0,1}
    const float* arow = &hls[(lane & 15) * LSTR + hi * 2];
    // B operand bases per gate (coalesced float2 per lane, +64 floats per ks)
    const float* bg0 = ws + (((0 * NT + wave) * KS) * 32 + lane) * 2;
    const float* bg1 = ws + (((1 * NT + wave) * KS) * 32 + lane) * 2;
    const float* bg2 = ws + (((2 * NT + wave) * KS) * 32 + lane) * 2;
    const float* bg3 = ws + (((3 * NT + wave) * KS) * 32 + lane) * 2;

    // per-lane x row pointer: rows m0 + hi*8 + r
    const float* xrow = x + (m0 + hi * 8) * SEQ;

    v8f cst = {};                             // cell state, C/D layout

    __syncthreads();

    for (int s = 0; s < SEQ; ++s) {
        // hoist the 8 x loads ahead of the WMMA phase: LOADcnt retires in
        // order, so they are complete by the time the last B-load is waited on
        float xr[8];
#pragma unroll
        for (int r = 0; r < 8; ++r) xr[r] = xrow[r * SEQ + s];

        v8f a0 = {}, a1 = {}, a2 = {}, a3 = {};

#pragma unroll 8
        for (int ks = 0; ks < KS; ++ks) {
            v2f av = *(const v2f*)(arow + ks * 4);     // shared A for 4 gates
            v2f b0 = *(const v2f*)(bg0 + ks * 64);
            v2f b1 = *(const v2f*)(bg1 + ks * 64);
            v2f b2 = *(const v2f*)(bg2 + ks * 64);
            v2f b3 = *(const v2f*)(bg3 + ks * 64);
            a0 = __builtin_amdgcn_wmma_f32_16x16x4_f32(false, av, false, b0,
                                                       (short)0, a0, false, false);
            a1 = __builtin_amdgcn_wmma_f32_16x16x4_f32(false, av, false, b1,
                                                       (short)0, a1, false, false);
            a2 = __builtin_amdgcn_wmma_f32_16x16x4_f32(false, av, false, b2,
                                                       (short)0, a2, false, false);
            a3 = __builtin_amdgcn_wmma_f32_16x16x4_f32(false, av, false, b3,
                                                       (short)0, a3, false, false);
        }

        // elementwise gates + state update; C/D layout: row m = hi*8 + r
        float hnew[8];
#pragma unroll
        for (int r = 0; r < 8; ++r) {
            const float xv = xr[r];
            const float gg = fast_tanh   (a0[r] + xv * wx0 + bb0);
            const float ii = fast_sigmoid(a1[r] + xv * wx1 + bb1);
            const float ff = fast_sigmoid(a2[r] + xv * wx2 + bb2);
            const float oo = fast_sigmoid(a3[r] + xv * wx3 + bb3);
            const float cc = gg * ii + cst[r] * ff;
            cst[r]  = cc;
            hnew[r] = fast_tanh(cc) * oo;
        }

        __syncthreads();                       // all A-reads done before overwrite
#pragma unroll
        for (int r = 0; r < 8; ++r)
            hls[(hi * 8 + r) * LSTR + n0 + nn] = hnew[r];
        __syncthreads();                       // h complete before next step
    }

    // final projection: out = h @ Wp + b_p   (bp buffer is zeros)
    if (tid < 160) {
        const int b = tid / 10, o = tid % 10;
        float acc = b_p[o];
        for (int k = 0; k < HDIM; ++k)
            acc += hls[b * LSTR + k] * Wp[k * 10 + o];
        out[(m0 + b) * 10 + o] = acc;
    }
}

// ---------------------------------------------------------------------------
extern "C" void kernel_launch(void* const* d_in, const int* in_sizes, int n_in,
                              void* d_out, int out_size, void* d_ws, size_t ws_size,
                              hipStream_t stream) {
    const float* x    = (const float*)d_in[0];
    const float* Wgx  = (const float*)d_in[1];
    const float* b_gx = (const float*)d_in[2];
    const float* Wgh  = (const float*)d_in[3];
    const float* b_gh = (const float*)d_in[4];
    const float* Wix  = (const float*)d_in[5];
    const float* b_ix = (const float*)d_in[6];
    const float* Wih  = (const float*)d_in[7];
    const float* b_ih = (const float*)d_in[8];
    const float* Wfx  = (const float*)d_in[9];
    const float* b_fx = (const float*)d_in[10];
    const float* Wfh  = (const float*)d_in[11];
    const float* b_fh = (const float*)d_in[12];
    const float* Wox  = (const float*)d_in[13];
    const float* b_ox = (const float*)d_in[14];
    const float* Woh  = (const float*)d_in[15];
    const float* b_oh = (const float*)d_in[16];
    const float* Wp   = (const float*)d_in[17];
    const float* b_p  = (const float*)d_in[18];
    float* ws  = (float*)d_ws;                 // needs 1 MB (4*16*64*64 floats)
    float* out = (float*)d_out;

    lstm_repack<<<512, 256, 0, stream>>>(Wgh, Wih, Wfh, Woh, ws);
    lstm_main<<<16, 512, 0, stream>>>(x, ws,
                                      Wgx, b_gx, b_gh,
                                      Wix, b_ix, b_ih,
                                      Wfx, b_fx, b_fh,
                                      Wox, b_ox, b_oh,
                                      Wp, b_p, out);
}